// FCPairedLayer_61323543052293
// MI455X (gfx1250) — compile-verified
//
#include <hip/hip_runtime.h>

typedef float v2f __attribute__((ext_vector_type(2)));
typedef float v8f __attribute__((ext_vector_type(8)));

#define BB 4
#define CC 64
#define NN 512
#define HH 128

// ---------------------------------------------------------------------------
// Kernel 1: L = xt * W1[:64,:],  R = xt * W1[64:,:] + b1   (fp32 WMMA)
// Combined GEMM: (B*N = 2048) x (K = 64) x (256 cols: [L | R]).
// One wave computes one 16x16 tile via 16 x v_wmma_f32_16x16x4_f32.
// Output layout (h-major): Lg[(b*H + h)*N + n], Rg likewise.
// ---------------------------------------------------------------------------
__global__ __launch_bounds__(256) void fcpaired_precompute_lr(
    const float* __restrict__ x,   // (B, C, N)
    const float* __restrict__ W1,  // (2C, H) row-major
    const float* __restrict__ b1,  // (H)
    float* __restrict__ Lg, float* __restrict__ Rg)
{
  const int wave = (blockIdx.x * 256 + threadIdx.x) >> 5;  // 0..2047
  const int lane = threadIdx.x & 31;
  const int tile_m = wave & 127;   // 128 tiles over B*N
  const int tile_h = wave >> 7;    // 16 tiles over 256 columns
  const int m0 = tile_m * 16;
  const int h0 = tile_h * 16;
  const int b  = m0 >> 9;          // uniform within tile (512 % 16 == 0)
  const int n0 = m0 & (NN - 1);
  const int half = lane >> 4;      // lane group (0: lanes 0-15, 1: 16-31)
  const int lr   = lane & 15;
  const int hc   = h0 + lr;        // 0..255 combined column
  const bool isR = hc >= HH;
  const int hcol = isR ? (hc - HH) : hc;

  v8f acc = {};
  #pragma unroll
  for (int s = 0; s < 16; ++s) {
    const int c0 = 4 * s + 2 * half;     // K base for this lane group
    // A (16x4 f32): VGPRv, lane-group g -> K = 2g+v, M = lane%16
    v2f a;
    a.x = x[((size_t)b * CC + c0 + 0) * NN + n0 + lr];
    a.y = x[((size_t)b * CC + c0 + 1) * NN + n0 + lr];
    // B (4x16 f32): same K striping; N = lane%16
    const int wrow = isR ? (CC + c0) : c0;
    v2f bm;
    bm.x = W1[(size_t)(wrow + 0) * HH + hcol];
    bm.y = W1[(size_t)(wrow + 1) * HH + hcol];
    acc = __builtin_amdgcn_wmma_f32_16x16x4_f32(
        /*neg_a=*/false, a, /*neg_b=*/false, bm,
        /*c_mod=*/(short)0, acc, /*reuse_a=*/false, /*reuse_b=*/false);
  }

  const float bias = isR ? b1[hcol] : 0.0f;
  float* __restrict__ dst = isR ? Rg : Lg;
  // D layout: VGPR r -> M = r + 8*half, N = lane%16
  #pragma unroll
  for (int r = 0; r < 8; ++r) {
    const int n = n0 + r + 8 * half;
    dst[((size_t)b * HH + hcol) * NN + n] = acc[r] + bias;
  }
}

// ---------------------------------------------------------------------------
// Kernel 2: score[b,i,j] = (j>i) ? sum_h relu(L[i,h]+R'[j,h]) * W2[h] + b2 : 0
// 64x64 output tile per 256-thread block; 4x4 register micro-tile per thread;
// L/R tiles staged h-major in LDS (64KB), ds_load_b128 per h step.
// ---------------------------------------------------------------------------
__global__ __launch_bounds__(256) void fcpaired_pairwise(
    const float* __restrict__ Lg, const float* __restrict__ Rg,
    const float* __restrict__ W2, const float* __restrict__ b2p,
    float* __restrict__ out)
{
  __shared__ __align__(16) float Lt[HH * 64];
  __shared__ __align__(16) float Rt[HH * 64];
  __shared__ float w2s[HH];

  const int tid = threadIdx.x;
  const int i0 = blockIdx.x * 64;
  const int j0 = blockIdx.y * 64;
  const int b  = blockIdx.z;
  const int tx = tid & 15;   // j direction
  const int ty = tid >> 4;   // i direction
  float* __restrict__ outB = out + (size_t)b * NN * NN;

  if (j0 + 63 <= i0) {  // entire tile below/at diagonal: zeros only
    const float4 z = {0.f, 0.f, 0.f, 0.f};
    #pragma unroll
    for (int ia = 0; ia < 4; ++ia) {
      const int i = i0 + ty * 4 + ia;
      *(float4*)(outB + (size_t)i * NN + j0 + tx * 4) = z;
    }
    return;
  }

  // Stage tiles: 2048 float4 per array, 8 per thread, coalesced rows.
  for (int t = tid; t < HH * 16; t += 256) {
    const int h = t >> 4, q = t & 15;
    ((float4*)Lt)[h * 16 + q] =
        *(const float4*)(Lg + ((size_t)b * HH + h) * NN + i0 + q * 4);
    ((float4*)Rt)[h * 16 + q] =
        *(const float4*)(Rg + ((size_t)b * HH + h) * NN + j0 + q * 4);
  }
  if (tid < HH) w2s[tid] = W2[tid];
  __syncthreads();

  float acc[4][4] = {};
  #pragma unroll 4
  for (int h = 0; h < HH; ++h) {
    const float4 lv = *(const float4*)(Lt + h * 64 + ty * 4);
    const float4 rv = *(const float4*)(Rt + h * 64 + tx * 4);
    const float w = w2s[h];
    const float l[4] = {lv.x, lv.y, lv.z, lv.w};
    const float r[4] = {rv.x, rv.y, rv.z, rv.w};
    #pragma unroll
    for (int ia = 0; ia < 4; ++ia)
      #pragma unroll
      for (int jb = 0; jb < 4; ++jb)
        acc[ia][jb] = __builtin_fmaf(fmaxf(l[ia] + r[jb], 0.0f), w, acc[ia][jb]);
  }

  const float b2 = b2p[0];
  #pragma unroll
  for (int ia = 0; ia < 4; ++ia) {
    const int i = i0 + ty * 4 + ia;
    float4 v;
    float res[4];
    #pragma unroll
    for (int jb = 0; jb < 4; ++jb) {
      const int j = j0 + tx * 4 + jb;
      res[jb] = (j > i) ? (acc[ia][jb] + b2) : 0.0f;
    }
    v.x = res[0]; v.y = res[1]; v.z = res[2]; v.w = res[3];
    *(float4*)(outB + (size_t)i * NN + j0 + tx * 4) = v;
  }
}

extern "C" void kernel_launch(void* const* d_in, const int* in_sizes, int n_in,
                              void* d_out, int out_size, void* d_ws, size_t ws_size,
                              hipStream_t stream) {
  const float* x  = (const float*)d_in[0];   // (4, 64, 512)
  const float* W1 = (const float*)d_in[1];   // (128, 128)
  const float* b1 = (const float*)d_in[2];   // (128)
  const float* W2 = (const float*)d_in[3];   // (128, 1)
  const float* b2 = (const float*)d_in[4];   // (1)
  float* out = (float*)d_out;                // (4, 512, 512)

  float* Lg = (float*)d_ws;                  // B*H*N floats = 1 MB
  float* Rg = Lg + (size_t)BB * HH * NN;     // + 1 MB

  // 2048 waves (one 16x16 WMMA tile each) = 256 blocks x 256 threads
  fcpaired_precompute_lr<<<256, 256, 0, stream>>>(x, W1, b1, Lg, Rg);

  dim3 grid(NN / 64, NN / 64, BB);           // (8, 8, 4)
  fcpaired_pairwise<<<grid, 256, 0, stream>>>(Lg, Rg, W2, b2, out);
}